// QuadConvLayer_34076270526769
// MI455X (gfx1250) — compile-verified
//
#include <hip/hip_runtime.h>

typedef float v2f __attribute__((ext_vector_type(2)));
typedef float v8f __attribute__((ext_vector_type(8)));

#define B_     4
#define C_IN_  32
#define C_OUT_ 32
#define N_IN_  4096
#define N_OUT_ 1024
#define E_     400000

#define NCHUNK_   64                       // edge chunks per batch
#define AGG_ELEMS (N_OUT_ * C_IN_)         // 32768 floats = 128KB (per batch)

// ---------------------------------------------------------------------------
// Phase 0: transpose features (B, C, N_IN) -> featT (B, N_IN, C)
// Writes coalesced (c innermost); reads strided but source is 2MB / L2-hot.
// ---------------------------------------------------------------------------
__global__ void k_transpose_feat(const float* __restrict__ f, float* __restrict__ ft) {
    int idx = blockIdx.x * blockDim.x + threadIdx.x;      // (b, n, c), c innermost
    if (idx >= B_ * N_IN_ * C_IN_) return;
    int c = idx & 31;
    int n = (idx >> 5) & (N_IN_ - 1);
    int b = idx >> 17;                                    // 5 + 12 bits
    ft[idx] = f[(b * C_IN_ + c) * N_IN_ + n];
}

// ---------------------------------------------------------------------------
// Phase 1: zero the global accumulator (ws is poisoned once; re-zero per call)
// ---------------------------------------------------------------------------
__global__ void k_zero_agg(float* __restrict__ agg) {
    int idx = blockIdx.x * blockDim.x + threadIdx.x;
    if (idx < B_ * AGG_ELEMS) agg[idx] = 0.0f;
}

// ---------------------------------------------------------------------------
// Phase 2: LDS-privatized scatter-add.
//   blockIdx.y = batch b, blockIdx.x = edge chunk.
//   Each block owns a private 128KB LDS accumulator (1024 x 32 f32) for its
//   batch, turns the 51.2M adds into ds_add_f32, then flushes once with
//   coalesced global atomics (256 blocks x 128KB = 32MB vs 204MB direct).
//   Waves load 32 edge indices coalesced, then broadcast via __shfl.
// ---------------------------------------------------------------------------
__global__ void k_scatter_lds(const float* __restrict__ ft,
                              const int*   __restrict__ idx_out,
                              const int*   __restrict__ idx_in,
                              float*       __restrict__ agg) {
    extern __shared__ float priv[];                  // AGG_ELEMS floats = 128KB
    const int b     = blockIdx.y;
    const int chunk = blockIdx.x;
    const int tid   = threadIdx.x;                   // 0..255 (8 waves)
    const int lane  = tid & 31;
    const int wave  = tid >> 5;

    // zero private accumulator
    for (int i = tid; i < AGG_ELEMS; i += 256) priv[i] = 0.0f;
    __syncthreads();

    const float* ftb = ft + (size_t)b * (N_IN_ * C_IN_);

    const int per_chunk = (E_ + NCHUNK_ - 1) / NCHUNK_;
    const int start = chunk * per_chunk;
    const int end   = min(E_, start + per_chunk);

    // each wave walks 32-edge groups; 8 waves interleave with stride 256
    for (int g = start + wave * 32; g < end; g += 8 * 32) {
        int e  = g + lane;
        int ii = 0, oo = 0;
        if (e < end) { ii = idx_in[e]; oo = idx_out[e]; }   // coalesced
        if (g + 32 <= end) {
#pragma unroll
            for (int j = 0; j < 32; ++j) {                  // v_readlane broadcast
                int iij = __shfl(ii, j, 32);
                int ooj = __shfl(oo, j, 32);
                float v = ftb[iij * C_IN_ + lane];           // 128B coalesced
                atomicAdd(&priv[ooj * C_IN_ + lane], v);     // ds_add_f32
            }
        } else {
            int cnt = end - g;
            for (int j = 0; j < cnt; ++j) {
                int iij = __shfl(ii, j, 32);
                int ooj = __shfl(oo, j, 32);
                float v = ftb[iij * C_IN_ + lane];
                atomicAdd(&priv[ooj * C_IN_ + lane], v);
            }
        }
    }
    __syncthreads();

    // flush: coalesced global atomics into agg[b]
    float* aggb = agg + (size_t)b * AGG_ELEMS;
    for (int i = tid; i < AGG_ELEMS; i += 256) {
        float v = priv[i];
        if (v != 0.0f) atomicAdd(&aggb[i], v);
    }
}

// ---------------------------------------------------------------------------
// Phase 3: out[b,o,n] = sum_k G[o,k] * agg[b,n,k]  via V_WMMA_F32_16X16X4_F32.
// One wave per 16x16 output tile; K=32 -> 8 chained WMMAs.
// ---------------------------------------------------------------------------
__global__ void k_gemm_wmma(const float* __restrict__ G,
                            const float* __restrict__ agg,
                            float* __restrict__ out) {
    int tile = blockIdx.x;            // 0..511
    int nt   = tile & 63;             // n-tile: 0..63
    int mt   = (tile >> 6) & 1;       // m-tile: 0..1
    int b    = tile >> 7;             // batch:  0..3

    int lane = threadIdx.x & 31;
    int half = lane >> 4;             // 0 or 1
    int l15  = lane & 15;

    int n0 = nt * 16;
    int o0 = mt * 16;

    const float* Ga   = G   + (o0 + l15) * C_IN_;               // A row (M = o)
    const float* aggB = agg + (b * N_OUT_ + n0 + l15) * C_IN_;  // B col (N = n)

    v8f acc = {};
#pragma unroll
    for (int k0 = 0; k0 < C_IN_; k0 += 4) {
        v2f a  = *(const v2f*)(Ga   + k0 + 2 * half);
        v2f bm = *(const v2f*)(aggB + k0 + 2 * half);
        acc = __builtin_amdgcn_wmma_f32_16x16x4_f32(
            /*neg_a=*/false, a, /*neg_b=*/false, bm,
            /*c_mod=*/(short)0, acc, /*reuse_a=*/false, /*reuse_b=*/false);
    }

    float* outp = out + (b * C_OUT_ + o0) * N_OUT_ + n0 + l15;
#pragma unroll
    for (int j = 0; j < 8; ++j) {
        outp[(j + 8 * half) * N_OUT_] = acc[j];
    }
}

// ---------------------------------------------------------------------------
extern "C" void kernel_launch(void* const* d_in, const int* in_sizes, int n_in,
                              void* d_out, int out_size, void* d_ws, size_t ws_size,
                              hipStream_t stream) {
    const float* features = (const float*)d_in[0];   // (4, 32, 4096) f32
    const float* G        = (const float*)d_in[1];   // (32, 32) f32
    const int*   idx_out  = (const int*)  d_in[2];   // (E,) int32
    const int*   idx_in   = (const int*)  d_in[3];   // (E,) int32
    float*       out      = (float*)d_out;           // (4, 32, 1024) f32

    // Workspace layout: [agg: 4*32768 f32 = 512KB][featT: 4*4096*32 f32 = 2MB]
    float* agg   = (float*)d_ws;
    float* featT = agg + (size_t)B_ * AGG_ELEMS;

    {   // zero accumulator
        int n = B_ * AGG_ELEMS;
        k_zero_agg<<<(n + 255) / 256, 256, 0, stream>>>(agg);
    }
    {   // transpose features
        int n = B_ * N_IN_ * C_IN_;
        k_transpose_feat<<<(n + 255) / 256, 256, 0, stream>>>(features, featT);
    }
    {   // LDS-privatized scatter-add: 64 chunks x 4 batches, 128KB dyn-LDS each
        const int lds_bytes = AGG_ELEMS * sizeof(float);     // 131072
        static bool attr_set = false;   // idempotent host-side attribute (capture-safe)
        if (!attr_set) {
            hipFuncSetAttribute((const void*)k_scatter_lds,
                                hipFuncAttributeMaxDynamicSharedMemorySize, lds_bytes);
            attr_set = true;
        }
        dim3 grid(NCHUNK_, B_);
        k_scatter_lds<<<grid, 256, lds_bytes, stream>>>(featT, idx_out, idx_in, agg);
    }
    {   // WMMA GEMM: 4 batches * 2 m-tiles * 64 n-tiles = 512 waves
        k_gemm_wmma<<<512, 32, 0, stream>>>(G, agg, out);
    }
}